// GIN_7507602834021
// MI455X (gfx1250) — compile-verified
//
#include <hip/hip_runtime.h>
#include <hip/hip_bf16.h>

// ---------------------------------------------------------------------------
// GIN (2x GINConv + FC + log_softmax), N=100000 nodes, E=1.6M edges, H=64, C=16
// Memory-bound (scatter-add dominates; feat/agg buffers are L2-resident).
// GEMMs fused per pass, run on V_WMMA_F32_16X16X4_F32 (fp32 numerics).
// Weights are stored in LDS in K-pair-interleaved layout so each lane's B
// fragment is ONE ds_load_b64 into an aligned VGPR pair (no v_mov packing).
// ---------------------------------------------------------------------------

typedef __attribute__((ext_vector_type(2))) float v2f;
typedef __attribute__((ext_vector_type(8))) float v8f;

#define ASTRIDE 66    // 16x64 A-tile row stride (dwords): bank-padded, rows 8B-aligned
#define WPAIRS  32    // 64 K values -> 32 K-pairs
#define WSTRIDE 160   // dwords per K-pair row (64 cols * 2) + 32 pad -> halves on disjoint banks
#define FSTRIDE 32    // fc: 16 cols * 2; offset 32 mod 64 -> already conflict-free

__device__ __forceinline__ v8f wmma4(v2f a, v2f b, v8f c) {
  // D = A(16x4,f32) * B(4x16,f32) + C(16x16,f32)
  return __builtin_amdgcn_wmma_f32_16x16x4_f32(
      /*neg_a=*/false, a, /*neg_b=*/false, b,
      /*c_mod=*/(short)0, c, /*reuse_a=*/false, /*reuse_b=*/false);
}

// ---------------------------------------------------------------------------
// Scatter-add aggregation: agg[dst[e]][f] += feat[src[e]][f]
// One thread per (edge, feature); f fast -> coalesced gather + coalesced
// global_atomic_add_f32 (each wave covers half an edge's feature row).
// ---------------------------------------------------------------------------
__global__ __launch_bounds__(256) void scatter_add_kernel(
    float* __restrict__ agg, const float* __restrict__ feat,
    const int* __restrict__ src, const int* __restrict__ dst, long long total) {
  long long idx = (long long)blockIdx.x * 256 + threadIdx.x;
  if (idx >= total) return;
  int e = (int)(idx >> 6);
  int f = (int)(idx & 63);
  int s = src[e];
  int d = dst[e];
  atomicAdd(&agg[(size_t)d * 64 + f], feat[(size_t)s * 64 + f]);
}

// ---- LDS helpers -----------------------------------------------------------

// 64x64 weights -> pair-interleaved: sW[q*WSTRIDE + n*2 + r] = W[2q+r][n]
__device__ __forceinline__ void load_weights64(const float* __restrict__ W, float* sW) {
  for (int i = threadIdx.x; i < 64 * 64; i += 256) {
    int k = i >> 6, n = i & 63;
    sW[(k >> 1) * WSTRIDE + n * 2 + (k & 1)] = W[i];
  }
}

// Stage one wave's 16x64 node tile (rows >= n zero-filled) into its LDS slab.
__device__ __forceinline__ void stage_tile(const float* __restrict__ hin,
                                           float* slab, int rowstart, int n, int lane) {
  const float4* g = (const float4*)hin;
#pragma unroll
  for (int i = 0; i < 8; ++i) {
    int t  = i * 32 + lane;   // 256 float4 slots = 16 rows x 16 float4
    int r  = t >> 4;
    int c4 = t & 15;
    float4 v = make_float4(0.f, 0.f, 0.f, 0.f);
    int row = rowstart + r;
    if (row < n) v = g[(size_t)row * 16 + c4];
    float* p = &slab[r * ASTRIDE + c4 * 4];
    p[0] = v.x; p[1] = v.y; p[2] = v.z; p[3] = v.w;
  }
}

// 16x64 tile (slab) x 64x64 weights (pair-interleaved sW): 16 K-steps x 4 N-tiles.
// A frag: ds_load_b64 from slab (8B-aligned: 264*col + 4*ka, ka even).
// B frag: ds_load_b64 from sW pair layout.
__device__ __forceinline__ void gemm_tile(const float* slab, const float* sW,
                                          v8f acc[4], int lane) {
  const int half = lane >> 4;
  const int col  = lane & 15;
#pragma unroll
  for (int kb = 0; kb < 64; kb += 4) {
    const int ka = kb + 2 * half;            // even; per-half K pair (ISA f32 layout)
    const int q  = ka >> 1;
    v2f a = *(const v2f*)(slab + col * ASTRIDE + ka);
#pragma unroll
    for (int nt = 0; nt < 4; ++nt) {
      const int nn = nt * 16 + col;
      v2f b = *(const v2f*)(sW + q * WSTRIDE + nn * 2);
      acc[nt] = wmma4(a, b, acc[nt]);
    }
  }
}

// D-layout (VGPR v -> row v+8*half, lane -> col) back into A-layout slab,
// with bias and optional ReLU. Same-wave LDS ops are in-order (DScnt).
__device__ __forceinline__ void result_to_slab(const v8f acc[4], const float* sB,
                                               float* slab, int lane, bool relu) {
  const int half = lane >> 4;
  const int col  = lane & 15;
#pragma unroll
  for (int nt = 0; nt < 4; ++nt) {
    const float bv = sB[nt * 16 + col];
#pragma unroll
    for (int v = 0; v < 8; ++v) {
      float x = acc[nt][v] + bv;
      if (relu) x = fmaxf(x, 0.f);
      slab[(v + 8 * half) * ASTRIDE + nt * 16 + col] = x;
    }
  }
}

// ---------------------------------------------------------------------------
// Layer 1: hout = relu( relu((x+agg)@Wa+ba) @ Wb + bb )  (outer relu = F.relu)
// Block = 8 waves; each wave owns a 16-row tile (128 rows/block).
// ---------------------------------------------------------------------------
__global__ __launch_bounds__(256) void gin_layer_relu_kernel(
    const float* __restrict__ hin,
    const float* __restrict__ Wa, const float* __restrict__ ba,
    const float* __restrict__ Wb, const float* __restrict__ bb,
    float* __restrict__ hout, int n) {
  __shared__ float sWa[WPAIRS * WSTRIDE];
  __shared__ float sWb[WPAIRS * WSTRIDE];
  __shared__ float sBa[64], sBb[64];
  __shared__ float slab[8][16 * ASTRIDE];

  load_weights64(Wa, sWa);
  load_weights64(Wb, sWb);
  if (threadIdx.x < 64) { sBa[threadIdx.x] = ba[threadIdx.x]; sBb[threadIdx.x] = bb[threadIdx.x]; }

  const int wave = threadIdx.x >> 5;
  const int lane = threadIdx.x & 31;
  const int rowstart = blockIdx.x * 128 + wave * 16;

  stage_tile(hin, slab[wave], rowstart, n, lane);
  __syncthreads();

  v8f acc[4] = {};
  gemm_tile(slab[wave], sWa, acc, lane);
  result_to_slab(acc, sBa, slab[wave], lane, /*relu=*/true);
  __syncthreads();

  v8f acc2[4] = {};
  gemm_tile(slab[wave], sWb, acc2, lane);

  const int half = lane >> 4;
  const int col  = lane & 15;
#pragma unroll
  for (int nt = 0; nt < 4; ++nt) {
    const float bv = sBb[nt * 16 + col];
#pragma unroll
    for (int v = 0; v < 8; ++v) {
      const int row = rowstart + v + 8 * half;
      if (row < n)
        hout[(size_t)row * 64 + nt * 16 + col] = fmaxf(acc2[nt][v] + bv, 0.f);
    }
  }
}

// ---------------------------------------------------------------------------
// Layer 2 fused with FC + log_softmax:
//   h2 = relu((h1+agg)@Wa+ba) @ Wb + bb        (no relu)
//   out = log_softmax(h2 @ Wfc + bfc)          (C = 16)
// Logit row r lives across one 16-lane half in accf[v]; reduce with shfl_xor.
// ---------------------------------------------------------------------------
__global__ __launch_bounds__(256) void gin_layer_fc_lsm_kernel(
    const float* __restrict__ hin,
    const float* __restrict__ Wa, const float* __restrict__ ba,
    const float* __restrict__ Wb, const float* __restrict__ bb,
    const float* __restrict__ Wfc, const float* __restrict__ bfc,
    float* __restrict__ out, int n) {
  __shared__ float sWa[WPAIRS * WSTRIDE];
  __shared__ float sWb[WPAIRS * WSTRIDE];
  __shared__ float sWf[WPAIRS * FSTRIDE];
  __shared__ float sBa[64], sBb[64], sBf[16];
  __shared__ float slab[8][16 * ASTRIDE];

  load_weights64(Wa, sWa);
  load_weights64(Wb, sWb);
  for (int i = threadIdx.x; i < 64 * 16; i += 256) {
    int k = i >> 4, c = i & 15;     // Wfc[k][c], row-major (64,16)
    sWf[(k >> 1) * FSTRIDE + c * 2 + (k & 1)] = Wfc[i];
  }
  if (threadIdx.x < 64) { sBa[threadIdx.x] = ba[threadIdx.x]; sBb[threadIdx.x] = bb[threadIdx.x]; }
  if (threadIdx.x < 16) sBf[threadIdx.x] = bfc[threadIdx.x];

  const int wave = threadIdx.x >> 5;
  const int lane = threadIdx.x & 31;
  const int rowstart = blockIdx.x * 128 + wave * 16;

  stage_tile(hin, slab[wave], rowstart, n, lane);
  __syncthreads();

  v8f acc[4] = {};
  gemm_tile(slab[wave], sWa, acc, lane);
  result_to_slab(acc, sBa, slab[wave], lane, /*relu=*/true);
  __syncthreads();

  v8f acc2[4] = {};
  gemm_tile(slab[wave], sWb, acc2, lane);
  result_to_slab(acc2, sBb, slab[wave], lane, /*relu=*/false);  // h2 tile back to slab
  __syncthreads();

  // FC: 16x64 tile @ 64x16 -> 16x16 logits (single N-tile, 16 K-steps)
  const int half = lane >> 4;
  const int col  = lane & 15;
  v8f accf = {};
#pragma unroll
  for (int kb = 0; kb < 64; kb += 4) {
    const int ka = kb + 2 * half;
    const int q  = ka >> 1;
    v2f a = *(const v2f*)(&slab[wave][col * ASTRIDE + ka]);
    v2f b = *(const v2f*)(sWf + q * FSTRIDE + col * 2);
    accf = wmma4(a, b, accf);
  }

  const float bv = sBf[col];
#pragma unroll
  for (int v = 0; v < 8; ++v) {
    float x = accf[v] + bv;
    // max over the 16 logits of this row (16-lane group reduction)
    float m = x;
#pragma unroll
    for (int mask = 1; mask < 16; mask <<= 1) m = fmaxf(m, __shfl_xor(m, mask, 16));
    float e = __expf(x - m);
    float s = e;
#pragma unroll
    for (int mask = 1; mask < 16; mask <<= 1) s += __shfl_xor(s, mask, 16);
    const float r = (x - m) - __logf(s);
    const int row = rowstart + v + 8 * half;
    if (row < n) out[(size_t)row * 16 + col] = r;
  }
}

// ---------------------------------------------------------------------------

extern "C" void kernel_launch(void* const* d_in, const int* in_sizes, int n_in,
                              void* d_out, int out_size, void* d_ws, size_t ws_size,
                              hipStream_t stream) {
  const float* x   = (const float*)d_in[0];
  const int*   ei  = (const int*)d_in[1];     // [2, E]: src row then dst row
  const float* W1a = (const float*)d_in[2];
  const float* b1a = (const float*)d_in[3];
  const float* W1b = (const float*)d_in[4];
  const float* b1b = (const float*)d_in[5];
  const float* W2a = (const float*)d_in[6];
  const float* b2a = (const float*)d_in[7];
  const float* W2b = (const float*)d_in[8];
  const float* b2b = (const float*)d_in[9];
  const float* Wfc = (const float*)d_in[10];
  const float* bfc = (const float*)d_in[11];

  const int n = in_sizes[0] / 64;
  const int e = in_sizes[1] / 2;
  const int* src = ei;
  const int* dst = ei + e;

  float* agg = (float*)d_ws;                  // n*64 : holds x(+agg), then h1(+agg)
  float* h1  = agg + (size_t)n * 64;          // n*64 : layer-1 output
  const size_t featBytes = (size_t)n * 64 * sizeof(float);

  const long long total = (long long)e * 64;
  const int sblocks = (int)((total + 255) / 256);
  const int lblocks = (n + 127) / 128;

  // Layer 1: agg = x; agg += scatter(x); h1 = MLP1(agg) with relu
  hipMemcpyAsync(agg, x, featBytes, hipMemcpyDeviceToDevice, stream);
  scatter_add_kernel<<<sblocks, 256, 0, stream>>>(agg, x, src, dst, total);
  gin_layer_relu_kernel<<<lblocks, 256, 0, stream>>>(agg, W1a, b1a, W1b, b1b, h1, n);

  // Layer 2 + FC + log_softmax fused
  hipMemcpyAsync(agg, h1, featBytes, hipMemcpyDeviceToDevice, stream);
  scatter_add_kernel<<<sblocks, 256, 0, stream>>>(agg, h1, src, dst, total);
  gin_layer_fc_lsm_kernel<<<lblocks, 256, 0, stream>>>(agg, W2a, b2a, W2b, b2b,
                                                       Wfc, bfc, (float*)d_out, n);
}